// crf_loss_func_87866440942112
// MI455X (gfx1250) — compile-verified
//
#include <hip/hip_runtime.h>
#include <hip/hip_bf16.h>
#include <math.h>

#define NB 64
#define SS 512
#define DD 768
#define CC 32

typedef float v2f __attribute__((ext_vector_type(2)));
typedef float v8f __attribute__((ext_vector_type(8)));

// ---------------------------------------------------------------------------
// Kernel A: npots[n,s,c] = sum_d X[n,s,d] * W[d,c]
// M = N*S = 32768 rows, K = 768, Ncols = 32.
// One wave handles a 16-row tile, producing two 16x16 WMMA accumulators
// (cols 0-15 and 16-31) via V_WMMA_F32_16X16X4_F32.
//
// f32 A 16x4 layout (ISA 7.12.2): lane = M (mod 16); VGPR0 holds K = 0|2
// (lane<16 | lane>=16), VGPR1 holds K = 1|3.  B 4x16 mirrors this with
// lane = N.  C/D: VGPR r -> row r (lanes 0-15) / r+8 (lanes 16-31).
// ---------------------------------------------------------------------------
__global__ __launch_bounds__(256) void gemm_npots_kernel(
    const float* __restrict__ X, const float* __restrict__ W,
    float* __restrict__ npots) {
  const int lane = threadIdx.x & 31;
  const int wave = threadIdx.x >> 5;
  const int rowTile = blockIdx.x * 8 + wave;       // 2048 tiles of 16 rows
  const int m  = lane & 15;                        // row / col within tile
  const int kh = (lane >> 4) << 1;                 // 0 or 2: K-half select
  const long row0 = (long)rowTile * 16;
  const float* __restrict__ xrow = X + (row0 + m) * DD;

  v8f acc0 = {0.f, 0.f, 0.f, 0.f, 0.f, 0.f, 0.f, 0.f};
  v8f acc1 = acc0;

  for (int k0 = 0; k0 < DD; k0 += 16) {
    if (k0 + 128 < DD) __builtin_prefetch(&xrow[k0 + 128], 0, 0);
#pragma unroll
    for (int kk = 0; kk < 16; kk += 4) {
      const int kb = k0 + kk;
      v2f a;
      a.x = xrow[kb + kh];
      a.y = xrow[kb + kh + 1];
      const float* wk = W + (long)(kb + kh) * CC;  // W row-major, stride 32
      v2f b0, b1;
      b0.x = wk[m];           b0.y = wk[CC + m];
      b1.x = wk[16 + m];      b1.y = wk[CC + 16 + m];
      acc0 = __builtin_amdgcn_wmma_f32_16x16x4_f32(
          false, a, false, b0, (short)0, acc0, false, false);
      acc1 = __builtin_amdgcn_wmma_f32_16x16x4_f32(
          false, a, false, b1, (short)0, acc1, false, false);
    }
  }

  const int mh = (lane >> 4) << 3;                 // 0 or 8
  float* out = npots + row0 * CC;
#pragma unroll
  for (int r = 0; r < 8; ++r) {
    out[(r + mh) * CC + m]      = acc0[r];
    out[(r + mh) * CC + 16 + m] = acc1[r];
  }
}

// ---------------------------------------------------------------------------
// Wave-level reductions (wave32)
// ---------------------------------------------------------------------------
__device__ __forceinline__ float wave_sum32(float v) {
#pragma unroll
  for (int off = 16; off > 0; off >>= 1) v += __shfl_xor(v, off, 32);
  return v;
}
__device__ __forceinline__ float wave_max32(float v) {
#pragma unroll
  for (int off = 16; off > 0; off >>= 1) v = fmaxf(v, __shfl_xor(v, off, 32));
  return v;
}

// ---------------------------------------------------------------------------
// Kernel B: per-batch backward CRF recursion + score + logZ.
// One wave per batch; lane c owns output class c and T row T[c,:] (32 VGPRs).
// Padded positions are handled purely by the mask (the reference's forward
// fill scan is provably dead under the masking structure).
// ---------------------------------------------------------------------------
__global__ __launch_bounds__(32) void crf_scan_kernel(
    const float* __restrict__ npots, const float* __restrict__ T,
    const float* __restrict__ labels, const float* __restrict__ pad_mask,
    float* __restrict__ partial) {
  const int n = blockIdx.x;
  const int c = threadIdx.x;                 // 0..31

  float Trow[CC];
#pragma unroll
  for (int j = 0; j < CC; ++j) Trow[j] = T[c * CC + j];

  __shared__ float v[CC];

  const float* np_n = npots    + (long)n * SS * CC;
  const float* lb_n = labels   + (long)n * SS * CC;
  const float* pm_n = pad_mask + (long)n * SS;

  float msg   = 0.0f;   // lane c holds msg[c]; starts at 0 beyond sequence end
  float emit  = 0.0f;   // per-lane emission-score partial
  float trans = 0.0f;   // uniform transition-score accumulator
  int   lab_next = 0;
  float m_next   = 0.0f;

  for (int t = SS - 1; t >= 0; --t) {
    const float mt  = pm_n[t];
    const float npc = np_n[t * CC + c];
    const float lbc = lb_n[t * CC + c];

    emit += npc * lbc * mt;

    // label index at t from one-hot: sum_c labels[c]*c
    const float labf = wave_sum32(lbc * (float)c);
    const int lab_t = (int)(labf + 0.5f);
    trans += T[lab_t * CC + lab_next] * m_next;   // pair (t, t+1), mask m_{t+1}

    if (t >= 1) {
      v[c] = npc + msg;                           // npots[t,c] + msg_t[c]
      __syncthreads();
      float x[CC];
      float mmax = -3.402823466e38f;
#pragma unroll
      for (int j = 0; j < CC; ++j) {
        x[j] = v[j] + Trow[j];
        mmax = fmaxf(mmax, x[j]);
      }
      float s = 0.0f;
#pragma unroll
      for (int j = 0; j < CC; ++j) s += __expf(x[j] - mmax);
      __syncthreads();
      msg = (mmax + __logf(s)) * mt;              // msg_{t-1}[c]
    }
    lab_next = lab_t;
    m_next   = mt;
  }

  // logZ = logsumexp_c(npots[0,c] + bwd0[c])
  const float z  = np_n[c] + msg;
  const float zm = wave_max32(z);
  const float zs = wave_sum32(__expf(z - zm));
  const float logZ = zm + __logf(zs);

  const float score = wave_sum32(emit) + trans;
  if (c == 0) partial[n] = logZ - score;
}

// ---------------------------------------------------------------------------
// Kernel C: deterministic final reduction, loss = mean(logZ - score)
// ---------------------------------------------------------------------------
__global__ __launch_bounds__(32) void final_reduce_kernel(
    const float* __restrict__ partial, float* __restrict__ out) {
  const int c = threadIdx.x;
  float s = partial[c] + partial[c + 32];
  s = wave_sum32(s);
  if (c == 0) out[0] = s * (1.0f / (float)NB);
}

// ---------------------------------------------------------------------------
extern "C" void kernel_launch(void* const* d_in, const int* in_sizes, int n_in,
                              void* d_out, int out_size, void* d_ws,
                              size_t ws_size, hipStream_t stream) {
  const float* X        = (const float*)d_in[0];
  const float* W        = (const float*)d_in[1];
  const float* T        = (const float*)d_in[2];
  const float* labels   = (const float*)d_in[3];
  const float* pad_mask = (const float*)d_in[4];

  float* npots   = (float*)d_ws;                                   // 4 MB
  float* partial = (float*)((char*)d_ws +
                            (size_t)NB * SS * CC * sizeof(float)); // 64 floats

  // 32768 rows / (16 rows/wave * 8 waves/block) = 256 blocks
  gemm_npots_kernel<<<(NB * SS) / (16 * 8), 256, 0, stream>>>(X, W, npots);
  crf_scan_kernel<<<NB, 32, 0, stream>>>(npots, T, labels, pad_mask, partial);
  final_reduce_kernel<<<1, 32, 0, stream>>>(partial, (float*)d_out);
}